// MyGAT2_3856880632471
// MI455X (gfx1250) — compile-verified
//
#include <hip/hip_runtime.h>

#define NN 23

typedef __attribute__((ext_vector_type(2))) float v2f;
typedef __attribute__((ext_vector_type(8))) float v8f;

// One wave per (einsum e, batch b, M-tile t).
// Attention row-normalization computed inline (branchless), then a 23x23
// matvec via 6 chained v_wmma_f32_16x16x4_f32 (K padded 23->24, M padded
// 23->32 over two 16-row tiles). The B operand carries the data vector
// broadcast across all 16 N columns, so every column of D holds the result.
__global__ void __launch_bounds__(128)
gat_wmma_kernel(const float* __restrict__ flow,   // data[j][i] = flow[j*23+i]
                const int*   __restrict__ adj,
                const float* __restrict__ W1,
                const float* __restrict__ W2,
                float* __restrict__ Y)            // Y[e*529 + b*23 + q]
{
    const int b    = blockIdx.x;          // batch index 0..22
    const int wave = threadIdx.x >> 5;    // 0..3
    const int lane = threadIdx.x & 31;
    const int e    = wave >> 1;           // 0: x1 einsum (W1), 1: x2 einsum (W2)
    const int t    = wave & 1;            // M tile (rows 0-15 / 16-31)
    const int hi   = lane >> 4;           // lane half selects K sub-pair
    const int m    = lane & 15;
    const int q    = t * 16 + m;          // output row owned by this lane pair
    const bool qv  = (q < NN);
    const int  qc  = qv ? q : 0;          // clamped row index (always in-bounds)

    const float* __restrict__ W    = (e == 0) ? W1 : W2;
    const float* __restrict__ Wrow = W + (b * NN * NN + qc * NN);
    const int*   __restrict__ Arow = adj + qc * NN;

    // data vector for this task, as base + stride (branchless):
    //   e==0: v[k] = data[k, b] = flow[k*23 + b]   -> base flow+b,    stride 23
    //   e==1: v[k] = data[b, k] = flow[b*23 + k]   -> base flow+b*23, stride 1
    const float* __restrict__ vbase = flow + ((e == 0) ? b : b * NN);
    const int vstride = (e == 0) ? NN : 1;

    // ---- attention row statistics (redundant across the two lane halves) ----
    // mn = min(min_j W[b,q,j], 0)  -- min over FULL row (mask not applied)
    float mn = 0.0f;
    #pragma unroll
    for (int j = 0; j < NN; ++j) mn = fminf(mn, Wrow[j]);

    // ssum = sum_j mask(q,j) ? (W - mn) : 0 ;  mask(q,j) = adj[q,j]!=0 || q==j
    float ssum = 0.0f;
    #pragma unroll
    for (int j = 0; j < NN; ++j) {
        const bool msk = (Arow[j] != 0) || (qc == j);
        ssum += msk ? (Wrow[j] - mn) : 0.0f;
    }
    const float inv = 1.0f / ssum;   // one divide per row (vs per element)

    // ---- batched matvec via WMMA f32 16x16x4, chained accumulator ----
    v8f acc = {0.f, 0.f, 0.f, 0.f, 0.f, 0.f, 0.f, 0.f};

    #pragma unroll
    for (int s = 0; s < 6; ++s) {
        const int kbase = 4 * s + 2 * hi;   // this lane's K pair within the step

        v2f afrag;
        v2f bfrag;
        #pragma unroll
        for (int v = 0; v < 2; ++v) {
            const int  k  = kbase + v;
            const bool kv = (k < NN);
            const int  kc = kv ? k : 0;     // clamped col index (always in-bounds)

            // unconditional loads from clamped (in-bounds) addresses,
            // then value selects (v_cndmask) -- no control flow.
            const float wv   = Wrow[kc];
            const bool  msk  = (Arow[kc] != 0) || (qc == kc);
            const float sel  = (qv && kv && msk) ? inv : 0.0f;
            const float av   = (wv - mn) * sel;

            const float dv   = vbase[kc * vstride];
            const float bv   = kv ? dv : 0.0f;

            if (v == 0) { afrag.x = av; bfrag.x = bv; }
            else        { afrag.y = av; bfrag.y = bv; }
        }

        // D = A(16x4) * B(4x16) + C
        acc = __builtin_amdgcn_wmma_f32_16x16x4_f32(
                  false, afrag, false, bfrag, (short)0, acc, false, false);
    }

    // ---- write out column n==0 (lanes 0 and 16): D[m = j + 8*hi, 0] ----
    if (m == 0) {
        #pragma unroll
        for (int j = 0; j < 8; ++j) {
            const int qq = t * 16 + j + 8 * hi;
            if (qq < NN) Y[e * NN * NN + b * NN + qq] = acc[j];
        }
    }
}

// out[n,i] = 0.5 * ( x1[n,i] + x2[n,i] ) = 0.5 * ( Y[0][i][n] + Y[1][n][i] )
__global__ void gat_combine(const float* __restrict__ Y, float* __restrict__ out) {
    const int idx = blockIdx.x * blockDim.x + threadIdx.x;
    if (idx < NN * NN) {
        const int n = idx / NN;
        const int i = idx % NN;
        out[idx] = 0.5f * (Y[0 * NN * NN + i * NN + n] + Y[1 * NN * NN + n * NN + i]);
    }
}

extern "C" void kernel_launch(void* const* d_in, const int* in_sizes, int n_in,
                              void* d_out, int out_size, void* d_ws, size_t ws_size,
                              hipStream_t stream) {
    const float* flow = (const float*)d_in[0];  // (1,23,23,1)
    const int*   adj  = (const int*)  d_in[1];  // (23,23)
    const float* W1   = (const float*)d_in[2];  // (23,23,23)
    const float* W2   = (const float*)d_in[3];  // (23,23,23)
    float* Y = (float*)d_ws;                    // 2*529 floats scratch

    gat_wmma_kernel<<<NN, 128, 0, stream>>>(flow, adj, W1, W2, Y);
    gat_combine<<<(NN * NN + 255) / 256, 256, 0, stream>>>(Y, (float*)d_out);
}